// Domain_Attention_67577015435620
// MI455X (gfx1250) — compile-verified
//
#include <hip/hip_runtime.h>
#include <math.h>

#define B_   128
#define N_   196
#define C_   768
#define CTR  64
#define NN   38416    // N_*N_
#define XS   150528   // N_*C_
#define TLSW 772      // padded LDS row stride (772 % 64 = 4 -> conflict-free column reads)

typedef __attribute__((ext_vector_type(2))) float v2f;
typedef __attribute__((ext_vector_type(8))) float v8f;

__device__ __forceinline__ v8f wmma4(v2f a, v2f b, v8f c) {
    // D = A(16x4,f32) * B(4x16,f32) + C(16x16,f32)  -> v_wmma_f32_16x16x4_f32
    return __builtin_amdgcn_wmma_f32_16x16x4_f32(
        /*neg_a=*/false, a, /*neg_b=*/false, b,
        /*c_mod=*/(short)0, c, /*reuse_a=*/false, /*reuse_b=*/false);
}

// ---------------- Stage 1: q = mean_N(x), L2-normalized -> qn (B,C) -------------
__global__ void qmean_norm_k(const float* __restrict__ x, float* __restrict__ qn) {
    int b = blockIdx.x, t = threadIdx.x;          // 256 threads, 3 channels each
    const float* xb = x + b * XS;
    float s0 = 0.f, s1 = 0.f, s2 = 0.f;
    for (int n = 0; n < N_; ++n) {
        const float* r = xb + n * C_;
        s0 += r[t]; s1 += r[t + 256]; s2 += r[t + 512];
    }
    const float inv = 1.0f / (float)N_;
    s0 *= inv; s1 *= inv; s2 *= inv;
    __shared__ float red[256];
    red[t] = s0 * s0 + s1 * s1 + s2 * s2;
    __syncthreads();
    for (int off = 128; off > 0; off >>= 1) {
        if (t < off) red[t] += red[t + off];
        __syncthreads();
    }
    float rinv = 1.0f / fmaxf(sqrtf(red[0]), 1e-12f);
    qn[b * C_ + t]       = s0 * rinv;
    qn[b * C_ + t + 256] = s1 * rinv;
    qn[b * C_ + t + 512] = s2 * rinv;
}

// ---------------- Stage 2: L2-normalize columns of k_c -> kn (C,CTR) -----------
__global__ void kc_norm_k(const float* __restrict__ kc, float* __restrict__ kn) {
    int j = blockIdx.x, t = threadIdx.x;
    float a0 = kc[t * CTR + j], a1 = kc[(t + 256) * CTR + j], a2 = kc[(t + 512) * CTR + j];
    __shared__ float red[256];
    red[t] = a0 * a0 + a1 * a1 + a2 * a2;
    __syncthreads();
    for (int off = 128; off > 0; off >>= 1) {
        if (t < off) red[t] += red[t + off];
        __syncthreads();
    }
    float rinv = 1.0f / fmaxf(sqrtf(red[0]), 1e-12f);
    kn[t * CTR + j]         = a0 * rinv;
    kn[(t + 256) * CTR + j] = a1 * rinv;
    kn[(t + 512) * CTR + j] = a2 * rinv;
}

// ---------------- Stage 3: attn = softmax((qn @ kn) * C^-0.5) -> (B,CTR) -------
__global__ void attn_k(const float* __restrict__ qn, const float* __restrict__ kn,
                       float* __restrict__ attn) {
    int b = blockIdx.x, j = threadIdx.x;          // 64 threads = 2 waves
    const float* q = qn + b * C_;
    float d = 0.f;
    for (int c = 0; c < C_; ++c) d += q[c] * kn[c * CTR + j];
    d *= 0.03608439182435161f;                    // 768^-0.5
    __shared__ float sm[CTR];
    sm[j] = d;
    __syncthreads();
    float mx = -1e30f;
    for (int i = 0; i < CTR; ++i) mx = fmaxf(mx, sm[i]);
    float e = __expf(d - mx);
    __syncthreads();
    sm[j] = e;
    __syncthreads();
    float s = 0.f;
    for (int i = 0; i < CTR; ++i) s += sm[i];
    attn[b * CTR + j] = e / s;
}

// ---------------- Stage 4a: pwt[k*C+c] = proj_w[c*C+k] (one-time, 2.4MB) -------
__global__ void transpose_k(const float* __restrict__ w, float* __restrict__ wt) {
    int idx = blockIdx.x * 256 + threadIdx.x;     // idx = k*C + c
    int k = idx / C_, c = idx % C_;
    wt[idx] = w[c * C_ + k];
}

// ---------------- Stage 4b: v = attn @ v_c  (M=128,K=64,N=38416) via WMMA ------
__global__ void __launch_bounds__(128) vmix_k(const float* __restrict__ attn,
                                              const float* __restrict__ vc,
                                              float* __restrict__ v) {
    int lane = threadIdx.x & 31;
    int wave = threadIdx.x >> 5;
    int tIdx = blockIdx.x * 4 + wave;             // N-tile per wave (uniform in wave)
    if (tIdx >= NN / 16) return;                  // 2401 tiles
    int hl   = lane & 15;
    int kb   = 2 * (lane >> 4);
    int mrow = blockIdx.y * 16 + hl;              // batch row (A layout)
    int ncol = tIdx * 16 + hl;                    // output column (B/C layout)
    v8f acc = {};
    for (int ks = 0; ks < 16; ++ks) {             // K = 64 = 16 * 4
        int k = ks * 4 + kb;
        v2f a, bv;
        a.x  = attn[mrow * CTR + k];
        a.y  = attn[mrow * CTR + k + 1];
        bv.x = vc[k * NN + ncol];
        bv.y = vc[(k + 1) * NN + ncol];
        acc = wmma4(a, bv, acc);
    }
    int rb = blockIdx.y * 16 + 8 * (lane >> 4);   // C layout: VGPR i -> M = i + 8*(lane>>4)
#pragma unroll
    for (int i = 0; i < 8; ++i) v[(rb + i) * NN + ncol] = acc[i];
}

// ------- Stage 5 (fused): out[b] = (v[b] @ x[b]) @ proj_w^T + proj_b -----------
// Block = 16 output rows (n) x full 768 cols; t-tile (16x768) lives in LDS.
__global__ void __launch_bounds__(128) outproj_k(const float* __restrict__ v,
                                                 const float* __restrict__ x,
                                                 const float* __restrict__ pwt,
                                                 const float* __restrict__ pb,
                                                 float* __restrict__ out) {
    int b    = blockIdx.y;
    int nt   = blockIdx.x;                         // 13 row tiles (last partial)
    int lane = threadIdx.x & 31;
    int wave = threadIdx.x >> 5;
    int hl   = lane & 15;
    int kb   = 2 * (lane >> 4);
    __shared__ float tls[16 * TLSW];               // 49,408 B of the 320KB WGP LDS
    const float* vb = v + b * NN;
    const float* xb = x + b * XS;
    int  r      = nt * 16 + hl;
    bool rvalid = r < N_;                          // lane-select, no branch around WMMA
    // ---- stage A: t_tile = v_rows(16x196) @ x[b](196x768), K = 196 = 49*4 ----
    for (int ct = wave; ct < 48; ct += 4) {        // 48 column tiles of 16
        int ccol = ct * 16 + hl;
        v8f acc = {};
        for (int ks = 0; ks < 49; ++ks) {
            int k = ks * 4 + kb;
            v2f a, bv;
            a.x  = rvalid ? vb[r * N_ + k]     : 0.0f;
            a.y  = rvalid ? vb[r * N_ + k + 1] : 0.0f;
            bv.x = xb[k * C_ + ccol];
            bv.y = xb[(k + 1) * C_ + ccol];
            acc = wmma4(a, bv, acc);
        }
        int rr = 8 * (lane >> 4);
#pragma unroll
        for (int i = 0; i < 8; ++i) tls[(rr + i) * TLSW + ccol] = acc[i];
    }
    __syncthreads();
    // ---- stage B: out_tile = t_tile(16x768) @ pwt(768x768), K = 768 = 192*4 ----
    for (int ct = wave; ct < 48; ct += 4) {
        int ccol = ct * 16 + hl;
        v8f acc = {};
        for (int ks = 0; ks < 192; ++ks) {
            int k = ks * 4 + kb;
            v2f a, bv;
            a.x  = tls[hl * TLSW + k];             // conflict-free: stride 772
            a.y  = tls[hl * TLSW + k + 1];
            bv.x = pwt[k * C_ + ccol];             // coalesced, L2-resident
            bv.y = pwt[(k + 1) * C_ + ccol];
            acc = wmma4(a, bv, acc);
        }
        float bias = pb[ccol];
        int rowb = nt * 16 + 8 * (lane >> 4);
#pragma unroll
        for (int i = 0; i < 8; ++i) {
            int row = rowb + i;
            if (row < N_) out[b * XS + row * C_ + ccol] = acc[i] + bias;
        }
    }
}

extern "C" void kernel_launch(void* const* d_in, const int* in_sizes, int n_in,
                              void* d_out, int out_size, void* d_ws, size_t ws_size,
                              hipStream_t stream) {
    const float* x  = (const float*)d_in[0];
    const float* kc = (const float*)d_in[1];
    const float* vc = (const float*)d_in[2];
    const float* pw = (const float*)d_in[3];
    const float* pb = (const float*)d_in[4];
    float* out = (float*)d_out;

    // workspace layout (floats): total ~22.7 MB
    float* ws   = (float*)d_ws;
    float* qn   = ws;                   // 128*768
    float* kn   = qn + B_ * C_;         // 768*64
    float* attn = kn + C_ * CTR;        // 128*64
    float* pwt  = attn + B_ * CTR;      // 768*768
    float* vws  = pwt + C_ * C_;        // 128*38416

    qmean_norm_k<<<B_, 256, 0, stream>>>(x, qn);
    kc_norm_k<<<CTR, 256, 0, stream>>>(kc, kn);
    attn_k<<<B_, CTR, 0, stream>>>(qn, kn, attn);
    transpose_k<<<(C_ * C_) / 256, 256, 0, stream>>>(pw, pwt);
    vmix_k<<<dim3((NN / 16 + 3) / 4, B_ / 16), 128, 0, stream>>>(attn, vc, vws);
    outproj_k<<<dim3((N_ + 15) / 16, B_), 128, 0, stream>>>(vws, x, pwt, pb, out);
}